// ReEig_42898133352587
// MI455X (gfx1250) — compile-verified
//
#include <hip/hip_runtime.h>

typedef __attribute__((ext_vector_type(2))) float v2f;
typedef __attribute__((ext_vector_type(8))) float v8f;
typedef int v4i_ __attribute__((vector_size(16)));   // matches async-LDS builtin's pointee type

#define NDIM  64
#define LSTR  68          // stride ≡ 4 (mod 64): paired-dword LDS reads cover all 64 banks once
#define EPSV  1e-4f
#define NITER 24          // Newton-Schulz sign iterations (1.5^k growth from ~1e-4 + quadratic tail)

#if __has_builtin(__builtin_amdgcn_global_load_async_to_lds_b128) && \
    __has_builtin(__builtin_amdgcn_s_wait_asynccnt)
#define HAS_ASYNC_LDS 1
#else
#define HAS_ASYNC_LDS 0
#endif

// 2x2 block of 16x16 output tiles of a 64x64 @ 64x64 fp32 GEMM out of LDS.
// B is SYMMETRIC here (all operands in this algorithm are), so the B fragment
// is read transposed -> contiguous K-pairs, same access shape as A.
__device__ __forceinline__ void gemm22(const float* sA, const float* sB,
                                       int mi0, int nj0, v8f c[2][2]) {
  const int lane = threadIdx.x & 31;
  const int half = lane >> 4;       // 0: K pair {0,1}, 1: K pair {2,3}
  const int l16  = lane & 15;
  const int ar0 = (mi0 * 16 + l16) * LSTR;
  const int ar1 = ar0 + 16 * LSTR;
  const int br0 = (nj0 * 16 + l16) * LSTR;   // transposed read: row n, cols k
  const int br1 = br0 + 16 * LSTR;
#pragma unroll
  for (int k0 = 0; k0 < NDIM; k0 += 4) {
    const int ka = k0 + half * 2;
    v2f a0 = *(const v2f*)&sA[ar0 + ka];
    v2f a1 = *(const v2f*)&sA[ar1 + ka];
    v2f b0 = *(const v2f*)&sB[br0 + ka];
    v2f b1 = *(const v2f*)&sB[br1 + ka];
    c[0][0] = __builtin_amdgcn_wmma_f32_16x16x4_f32(false, a0, false, b0, (short)0, c[0][0], false, false);
    c[0][1] = __builtin_amdgcn_wmma_f32_16x16x4_f32(false, a0, false, b1, (short)0, c[0][1], false, false);
    c[1][0] = __builtin_amdgcn_wmma_f32_16x16x4_f32(false, a1, false, b0, (short)0, c[1][0], false, false);
    c[1][1] = __builtin_amdgcn_wmma_f32_16x16x4_f32(false, a1, false, b1, (short)0, c[1][1], false, false);
  }
}

__device__ __forceinline__ void store22(float* sD, int mi0, int nj0, v8f c[2][2]) {
  const int lane = threadIdx.x & 31;
  const int half = lane >> 4;
  const int l16  = lane & 15;
#pragma unroll
  for (int i = 0; i < 2; ++i)
#pragma unroll
    for (int j = 0; j < 2; ++j)
#pragma unroll
      for (int r = 0; r < 8; ++r) {
        const int row = (mi0 + i) * 16 + r + half * 8;   // C/D layout: M=r / M=r+8 split at lane 16
        sD[row * LSTR + (nj0 + j) * 16 + l16] = c[i][j][r];
      }
}

__global__ __launch_bounds__(128)
void reeig_sign_kernel(const float* __restrict__ P, float* __restrict__ X) {
  __shared__ float sP[NDIM * LSTR];  // P0 = P - eps*I (symmetric)
  __shared__ float sY[NDIM * LSTR];  // sign iterate
  __shared__ float sT[NDIM * LSTR];  // temp (Y*Y)
  __shared__ float red[128];

  const int t = threadIdx.x;
  const long long base = (long long)blockIdx.x * (NDIM * NDIM);

  // ---- stage P into LDS ----
#if HAS_ASYNC_LDS
#pragma unroll
  for (int i = 0; i < 8; ++i) {
    const int idx4 = t + i * 128;            // 16B chunk id
    const int r = idx4 >> 4;
    const int c = (idx4 & 15) * 4;
    __builtin_amdgcn_global_load_async_to_lds_b128(
        (__attribute__((address_space(1))) v4i_*)(P + base + (long long)idx4 * 4),
        (__attribute__((address_space(3))) v4i_*)&sP[r * LSTR + c], 0, 0);
  }
  __builtin_amdgcn_s_wait_asynccnt(0);
  __syncthreads();
  if (t < NDIM) sP[t * LSTR + t] -= EPSV;    // P0 = P - eps*I
  __syncthreads();
  float acc2 = 0.f;
#pragma unroll
  for (int i = 0; i < 32; ++i) {
    const int idx = t + i * 128;
    const float v = sP[(idx >> 6) * LSTR + (idx & 63)];
    acc2 += v * v;
  }
#else
  float acc2 = 0.f;
#pragma unroll
  for (int i = 0; i < 32; ++i) {
    const int idx = t + i * 128;             // coalesced
    const int r = idx >> 6, c = idx & 63;
    float v = P[base + idx];
    if (r == c) v -= EPSV;
    sP[r * LSTR + c] = v;
    acc2 += v * v;
  }
#endif

  // ---- Frobenius norm -> Y0 = P0 / ||P0||_F ----
  red[t] = acc2;
  __syncthreads();
  for (int s = 64; s > 0; s >>= 1) {
    if (t < s) red[t] += red[t + s];
    __syncthreads();
  }
  const float inv_c = __frsqrt_rn(red[0] + 1e-30f);
#pragma unroll
  for (int i = 0; i < 32; ++i) {
    const int idx = t + i * 128;
    const int r = idx >> 6, c = idx & 63;
    sY[r * LSTR + c] = sP[r * LSTR + c] * inv_c;
  }
  __syncthreads();

  // ---- 4 waves, each owns a 2x2 block of 16x16 tiles ----
  const int wave = t >> 5;
  const int lane = t & 31;
  const int half = lane >> 4;
  const int l16  = lane & 15;
  const int mi0 = (wave >> 1) * 2;
  const int nj0 = (wave & 1) * 2;

  // Cache this lane's Y-tile elements (C/D-layout positions) in registers:
  // they are read and rewritten only by this lane across iterations.
  v8f y[2][2];
#pragma unroll
  for (int i = 0; i < 2; ++i)
#pragma unroll
    for (int j = 0; j < 2; ++j)
#pragma unroll
      for (int r = 0; r < 8; ++r) {
        const int row = (mi0 + i) * 16 + r + half * 8;
        y[i][j][r] = sY[row * LSTR + (nj0 + j) * 16 + l16];
      }

  // ---- Newton-Schulz: Y <- 1.5*Y - 0.5*Y*(Y*Y) ----
  for (int it = 0; it < NITER; ++it) {
    v8f c[2][2] = {};
    gemm22(sY, sY, mi0, nj0, c);             // T = Y*Y
    store22(sT, mi0, nj0, c);
    __syncthreads();                         // sT complete

    v8f z[2][2] = {};
    gemm22(sY, sT, mi0, nj0, z);             // Z = Y*T (kept in regs)
    __syncthreads();                         // all reads of sY done

#pragma unroll
    for (int i = 0; i < 2; ++i)
#pragma unroll
      for (int j = 0; j < 2; ++j)
#pragma unroll
        for (int r = 0; r < 8; ++r) {
          const int row = (mi0 + i) * 16 + r + half * 8;
          const float yn = 1.5f * y[i][j][r] - 0.5f * z[i][j][r];
          y[i][j][r] = yn;                   // register copy stays current
          sY[row * LSTR + (nj0 + j) * 16 + l16] = yn;
        }
    __syncthreads();
  }

  // ---- X = 0.5*(P0 + sign(P0)*P0) + eps*I ----
  v8f m[2][2] = {};
  gemm22(sY, sP, mi0, nj0, m);
#pragma unroll
  for (int i = 0; i < 2; ++i)
#pragma unroll
    for (int j = 0; j < 2; ++j)
#pragma unroll
      for (int r = 0; r < 8; ++r) {
        const int row = (mi0 + i) * 16 + r + half * 8;
        const int col = (nj0 + j) * 16 + l16;
        const float x = 0.5f * (sP[row * LSTR + col] + m[i][j][r]) + (row == col ? EPSV : 0.f);
        X[base + row * NDIM + col] = x;
      }
}

extern "C" void kernel_launch(void* const* d_in, const int* in_sizes, int n_in,
                              void* d_out, int out_size, void* d_ws, size_t ws_size,
                              hipStream_t stream) {
  (void)n_in; (void)out_size; (void)d_ws; (void)ws_size;
  const float* P = (const float*)d_in[0];
  float* X = (float*)d_out;
  const int nmat = in_sizes[0] / (NDIM * NDIM);   // 512*16 = 8192 matrices
  reeig_sign_kernel<<<nmat, 128, 0, stream>>>(P, X);
}